// SparseMAPWrapper_24309514895546
// MI455X (gfx1250) — compile-verified
//
#include <hip/hip_runtime.h>
#include <hip/hip_bf16.h>

// Problem sizes (fixed by the reference)
#define BB   8192   // batch
#define FF   1024   // input features
#define DD   128    // latent dim
#define DP1  129    // D+1
#define ROW_ELEMS (DP1 * DD)                 // 16512 floats per batch row of `sample`
#define SAMPLE_ELEMS ((size_t)BB * ROW_ELEMS)
#define DISTR_ELEMS  ((size_t)BB * DP1)

typedef float v2f __attribute__((ext_vector_type(2)));
typedef float v8f __attribute__((ext_vector_type(8)));
typedef float f4  __attribute__((ext_vector_type(4)));

#if __has_builtin(__builtin_amdgcn_wmma_f32_16x16x4_f32)
#define HAVE_WMMA_F32 1
#else
#define HAVE_WMMA_F32 0
#endif

// ---------------------------------------------------------------------------
// Kernel A: theta = x @ W + b, mu = clip(theta, 0, 1)  -> d_ws
// Block = 256 threads (8 waves). Each block computes a 16(M) x 128(N) tile;
// wave w owns the 16x16 N-subtile [16w,16w+16). K staged through LDS in
// chunks of 64; inner loop issues v_wmma_f32_16x16x4_f32.
// LDS strides padded (68 / 132 floats) for conflict-free fragment reads.
// ---------------------------------------------------------------------------
__global__ __launch_bounds__(256) void gemm_clip_kernel(
    const float* __restrict__ x, const float* __restrict__ W,
    const float* __restrict__ bias, float* __restrict__ mu)
{
  __shared__ __align__(16) float As[16][68];    // 16 x 64 (+pad)
  __shared__ __align__(16) float Bs[64][132];   // 64 x 128 (+pad)

  const int t    = threadIdx.x;
  const int wave = t >> 5;
  const int lane = t & 31;
  const int hi   = lane >> 4;     // 0: lanes 0-15, 1: lanes 16-31
  const int nl   = lane & 15;
  const int m0   = blockIdx.x * 16;
  const int n0   = wave * 16;

  v8f acc = {};                   // 16x16 fp32 accumulator tile (8 VGPRs)

  for (int k0 = 0; k0 < FF; k0 += 64) {
    // Stage A: 16 rows x 64 cols of x (one float4 per thread, coalesced)
    {
      const int r = t >> 4, c = (t & 15) << 2;
      const f4 v = *(const f4*)(x + (size_t)(m0 + r) * FF + k0 + c);
      As[r][c] = v.x; As[r][c + 1] = v.y; As[r][c + 2] = v.z; As[r][c + 3] = v.w;
    }
    // Stage B: 64 rows x 128 cols of W (8 float4 per thread, coalesced)
#pragma unroll
    for (int i = 0; i < 8; ++i) {
      const int idx = t + 256 * i;
      const int r = idx >> 5, c = (idx & 31) << 2;
      const f4 v = *(const f4*)(W + (size_t)(k0 + r) * DD + c);
      *(f4*)&Bs[r][c] = v;        // row stride 132 floats = 528 B, 16B-aligned
    }
    __syncthreads();

#if HAVE_WMMA_F32
#pragma unroll
    for (int kk = 0; kk < 64; kk += 4) {
      // A 16x4 f32 fragment: VGPR0 = K {0|2}, VGPR1 = K {1|3} (lo|hi half-wave)
      v2f a, b2;
      a.x  = As[nl][kk + 2 * hi];
      a.y  = As[nl][kk + 2 * hi + 1];
      // B 4x16 f32 fragment: rows striped across lanes (mirrors A striping)
      b2.x = Bs[kk + 2 * hi][n0 + nl];
      b2.y = Bs[kk + 2 * hi + 1][n0 + nl];
      acc = __builtin_amdgcn_wmma_f32_16x16x4_f32(
          /*neg_a=*/false, a, /*neg_b=*/false, b2,
          /*c_mod=*/(short)0, acc, /*reuse_a=*/false, /*reuse_b=*/false);
    }
#else
    // VALU fallback with identical C/D lane layout
#pragma unroll
    for (int i = 0; i < 8; ++i) {
      const int ml = i + 8 * hi;
      float s = 0.0f;
      for (int kk = 0; kk < 64; ++kk) s += As[ml][kk] * Bs[kk][n0 + nl];
      acc[i] += s;
    }
#endif
    __syncthreads();
  }

  // Epilogue: bias + clip, C/D layout: lane<16 -> M=i, lane>=16 -> M=i+8; N=n0+nl
  const int   n  = n0 + nl;
  const float bv = bias[n];
#pragma unroll
  for (int i = 0; i < 8; ++i) {
    const int m = m0 + i + 8 * hi;
    float v = acc[i] + bv;
    v = fminf(fmaxf(v, 0.0f), 1.0f);
    mu[(size_t)m * DD + n] = v;
  }
}

// ---------------------------------------------------------------------------
// Kernel B: per batch row -- rank vector g(d), sorted u via counting scatter,
// distr / entropy terms / support, then NT-stream the [129 x 128] sample tile.
// sample[b][j][d] = (j > g(d)) ? 1 : 0   (row j=0 is all zeros automatically)
// ---------------------------------------------------------------------------
__global__ __launch_bounds__(256) void rank_write_kernel(
    const float* __restrict__ mu, float* __restrict__ out,
    float* __restrict__ row_ent)
{
  const int b = blockIdx.x;
  const int t = threadIdx.x;

  __shared__ float smu[DD];
  __shared__ int   sg[DD];
  __shared__ float su[DD];
  __shared__ float sterm[DP1];
  __shared__ int   spos[DP1];

  if (t < DD) smu[t] = mu[(size_t)b * DD + t];
  __syncthreads();

  if (t < DD) {
    const float v = smu[t];
    int g = 0, e = 0;
#pragma unroll 8
    for (int d = 0; d < DD; ++d) {      // LDS broadcast reads, conflict-free
      const float ov = smu[d];
      g += (ov > v);
      e += ((d < t) && (ov == v));      // stable tie-break for the scatter
    }
    sg[t] = g;
    su[g + e] = v;                      // counting scatter -> descending sort
  }
  __syncthreads();

  // distr: w0 = 1-u0; w_j = u_{j-1}-u_j (j=1..127); w_128 = u_127
  float* distr = out + SAMPLE_ELEMS;
  if (t <= DD) {
    float w;
    if (t == 0)        w = 1.0f - su[0];
    else if (t < DD)   w = su[t - 1] - su[t];
    else               w = su[DD - 1];
    distr[(size_t)b * DP1 + t] = w;
    const bool p = (w > 0.0f);
    sterm[t] = p ? (-w * __logf(w)) : 0.0f;
    spos[t]  = p ? 1 : 0;
  }
  __syncthreads();

  // tree-reduce entries 0..127, fold in entry 128 at the end (deterministic)
  for (int s = 64; s > 0; s >>= 1) {
    if (t < s) { sterm[t] += sterm[t + s]; spos[t] += spos[t + s]; }
    __syncthreads();
  }
  if (t == 0) {
    row_ent[b] = sterm[0] + sterm[DD];
    float* support = out + SAMPLE_ELEMS + DISTR_ELEMS + 1;
    support[b] = (float)(spos[0] + spos[DD]);
  }

  // Stream the sample tile: each thread owns 4 consecutive d's (ranks hoisted
  // to registers), writes one 16B NT store per row; 8 rows per block pass.
  const int grp  = t >> 5;        // row group 0..7
  const int col4 = t & 31;        // float4 column within 128-float row
  const int d0   = col4 << 2;
  const int g0 = sg[d0], g1 = sg[d0 + 1], g2 = sg[d0 + 2], g3 = sg[d0 + 3];
  float* samp = out + (size_t)b * ROW_ELEMS + d0;
  for (int j = grp; j < DP1; j += 8) {
    f4 v;
    v.x = (j > g0) ? 1.0f : 0.0f;
    v.y = (j > g1) ? 1.0f : 0.0f;
    v.z = (j > g2) ? 1.0f : 0.0f;
    v.w = (j > g3) ? 1.0f : 0.0f;
    __builtin_nontemporal_store(v, (f4*)(samp + (size_t)j * DD));
  }
}

// ---------------------------------------------------------------------------
// Kernel C: deterministic fixed-order reduction of per-row entropies, /B
// ---------------------------------------------------------------------------
__global__ __launch_bounds__(256) void ent_reduce_kernel(
    const float* __restrict__ row_ent, float* __restrict__ out_ent)
{
  __shared__ float s[256];
  float acc = 0.0f;
  for (int i = threadIdx.x; i < BB; i += 256) acc += row_ent[i];
  s[threadIdx.x] = acc;
  __syncthreads();
  for (int r = 128; r > 0; r >>= 1) {
    if (threadIdx.x < (unsigned)r) s[threadIdx.x] += s[threadIdx.x + r];
    __syncthreads();
  }
  if (threadIdx.x == 0) *out_ent = s[0] * (1.0f / (float)BB);
}

// ---------------------------------------------------------------------------
extern "C" void kernel_launch(void* const* d_in, const int* in_sizes, int n_in,
                              void* d_out, int out_size, void* d_ws, size_t ws_size,
                              hipStream_t stream) {
  const float* x    = (const float*)d_in[0];   // [8192, 1024]
  const float* W    = (const float*)d_in[1];   // [1024, 128]
  const float* bias = (const float*)d_in[2];   // [128]
  float* out = (float*)d_out;

  // workspace: mu [8192*128] then row_entropy [8192]  (~4.2 MB)
  float* mu      = (float*)d_ws;
  float* row_ent = mu + (size_t)BB * DD;

  gemm_clip_kernel<<<BB / 16, 256, 0, stream>>>(x, W, bias, mu);
  rank_write_kernel<<<BB, 256, 0, stream>>>(mu, out, row_ent);
  ent_reduce_kernel<<<1, 256, 0, stream>>>(
      row_ent, out + SAMPLE_ELEMS + DISTR_ELEMS);
}